// LatencyPredictor_68788196213057
// MI455X (gfx1250) — compile-verified
//
#include <hip/hip_runtime.h>

typedef __attribute__((ext_vector_type(2))) float v2f;
typedef __attribute__((ext_vector_type(8))) float v8f;

#define BB   64
#define LX   8192
#define TT   64
#define LH   8129   /* LX - TT + 1 */
#define KK1  16
#define KK2  8
#define NTILE 509   /* ceil(8129/16) */

// ---------------------------------------------------------------------------
// Dendrite stage: conv (16x16 L-tile x K=64 via 16 f32 WMMAs) -> relu ->
// 16->8 GEMM (4 f32 WMMAs vs zero-padded W2) -> relu -> row stats.
// One wave (32 threads) per 16-row tile. blockIdx: (tile, batch, which A/B).
// ---------------------------------------------------------------------------
__global__ __launch_bounds__(32) void dendrite_kernel(
    const float* __restrict__ xa,  const float* __restrict__ xb,
    const float* __restrict__ W1Aw, const float* __restrict__ W1Ab,
    const float* __restrict__ W2Aw, const float* __restrict__ W2Ab,
    const float* __restrict__ W1Bw, const float* __restrict__ W1Bb,
    const float* __restrict__ W2Bw, const float* __restrict__ W2Bb,
    float* __restrict__ da,  float* __restrict__ ida,  float* __restrict__ vala,
    float* __restrict__ db,  float* __restrict__ idb,  float* __restrict__ valb,
    float* __restrict__ IaT, float* __restrict__ IbT)
{
    __shared__ float w1s[TT * KK1];   // 4 KB
    __shared__ float w2s[KK1 * 16];   // padded to 16 cols
    __shared__ float xw[96];          // 79 used (16 rows + 63 tail)
    __shared__ float d1t[16 * 16];
    __shared__ float d2t[16 * 16];

    const int lane  = threadIdx.x;
    const int tile  = blockIdx.x;
    const int b     = blockIdx.y;
    const int which = blockIdx.z;     // 0 = A, 1 = B
    const int l0    = tile * 16;

    const float* x   = which ? xb   : xa;
    const float* W1w = which ? W1Bw : W1Aw;
    const float* W1b = which ? W1Bb : W1Ab;
    const float* W2w = which ? W2Bw : W2Aw;
    const float* W2b = which ? W2Bb : W2Ab;
    float* out_d2  = which ? db   : da;
    float* out_idx = which ? idb  : ida;
    float* out_val = which ? valb : vala;
    float* I_T     = which ? IbT  : IaT;

    // Stage weights + x window in LDS
    for (int i = lane; i < TT * KK1; i += 32) w1s[i] = W1w[i];
    for (int i = lane; i < KK1 * 16; i += 32) {
        int k = i >> 4, j = i & 15;
        w2s[i] = (j < KK2) ? W2w[k * KK2 + j] : 0.0f;
    }
    for (int i = lane; i < 96; i += 32) {
        int gi = l0 + i;
        xw[i] = (i < (16 + TT - 1) && gi < LX) ? x[(size_t)b * LX + gi] : 0.0f;
    }
    __syncthreads();

    const int half = lane >> 4;   // 0: lanes 0-15, 1: lanes 16-31
    const int lc   = lane & 15;

    // GEMM 1: d1_pre[16x16] = H_tile[16x64] @ W1[64x16], f32 WMMA, K step = 4
    v8f acc = {};
#pragma unroll
    for (int kk = 0; kk < 16; ++kk) {
        v2f a, bm;
        // A 16x4 f32: lanes 0-15 hold K={4kk,4kk+1}, lanes 16-31 K={4kk+2,4kk+3}
        int ta = 4 * kk + 2 * half;
        a.x = xw[lc + ta];
        a.y = xw[lc + ta + 1];
        // B 4x16 f32: row striped across lanes; V0 rows {4kk,4kk+1}, V1 rows {+2,+3}
        int tb = 4 * kk + half;
        bm.x = w1s[tb * KK1 + lc];
        bm.y = w1s[(tb + 2) * KK1 + lc];
        acc = __builtin_amdgcn_wmma_f32_16x16x4_f32(
            false, a, false, bm, (short)0, acc, false, false);
    }

    // bias + relu; re-stripe C layout -> row-major LDS tile
    float bias1 = w1s[0], b1v;
    b1v = W1b[lc];
    (void)bias1;
#pragma unroll
    for (int r = 0; r < 8; ++r) {
        int row = r + 8 * half;          // C layout: VGPR r -> row r / r+8
        float v = acc[r] + b1v;
        d1t[row * 16 + lc] = v > 0.0f ? v : 0.0f;
    }
    __syncthreads();

    // GEMM 2: d2_pre[16x16] = d1[16x16] @ W2pad[16x16], 4 f32 WMMAs
    v8f acc2 = {};
#pragma unroll
    for (int kk = 0; kk < 4; ++kk) {
        v2f a, bm;
        int ta = 4 * kk + 2 * half;
        a.x = d1t[lc * 16 + ta];
        a.y = d1t[lc * 16 + ta + 1];
        int tb = 4 * kk + half;
        bm.x = w2s[tb * 16 + lc];
        bm.y = w2s[(tb + 2) * 16 + lc];
        acc2 = __builtin_amdgcn_wmma_f32_16x16x4_f32(
            false, a, false, bm, (short)0, acc2, false, false);
    }

    float b2v = (lc < KK2) ? W2b[lc] : 0.0f;
#pragma unroll
    for (int r = 0; r < 8; ++r) {
        int row = r + 8 * half;
        float v = acc2[r] + b2v;
        v = v > 0.0f ? v : 0.0f;
        d2t[row * 16 + lc] = v;
        int l = l0 + row;
        if (lc < KK2 && l < LH)
            out_d2[((size_t)b * LH + l) * KK2 + lc] = v;
    }
    __syncthreads();

    // Per-row stats: lanes 0-15, one row each
    if (half == 0) {
        int l = l0 + lc;
        if (l < LH) {
            float vmax = d2t[lc * 16 + 0];
            int imax = 0;
#pragma unroll
            for (int j = 1; j < KK2; ++j) {
                float v = d2t[lc * 16 + j];
                if (v > vmax) { vmax = v; imax = j; }
            }
            float esum = 0.0f, soma = 0.0f;
#pragma unroll
            for (int j = 0; j < KK2; ++j) {
                float v = d2t[lc * 16 + j];
                float e = __expf(v - vmax);   // shifted softmax == reference softmax
                esum += e;
                soma += e * v;
            }
            soma /= esum;
            size_t o = (size_t)b * LH + l;
            out_idx[o] = (float)imax;
            out_val[o] = vmax;
            I_T[(size_t)l * BB + b] = soma;   // transposed for coalesced LIF reads
        }
    }
}

// ---------------------------------------------------------------------------
// LIF scan: inherently sequential over LH; one lane per batch row, A and B
// halves in parallel blocks. Reads transposed I[t,b] -> fully coalesced.
// ---------------------------------------------------------------------------
__global__ __launch_bounds__(64) void lif_kernel(
    const float* __restrict__ IaT, const float* __restrict__ IbT,
    float* __restrict__ lah, float* __restrict__ las,
    float* __restrict__ lbh, float* __restrict__ lbs)
{
    const int which = blockIdx.x;
    const int b = threadIdx.x;
    const float* I_T = which ? IbT : IaT;
    float* out_hard  = which ? lbh : lah;
    float* out_soft  = which ? lbs : las;

    float v = 0.0f, smax = -1.0f, sum_s = 0.0f, sum_st = 0.0f;
    int imax = 0;
    for (int t = 0; t < LH; ++t) {
        float I = I_T[(size_t)t * BB + b];
        v = v - v * (1.0f / 20.0f) + I;                       // Euler step, TAU=20
        float s = 1.0f / (1.0f + __expf(-10.0f * (v - 1.0f))); // sigmoid(K*(v-TH))
        if (s > smax) { smax = s; imax = t; }                 // first argmax
        sum_s  += s;
        sum_st += s * (float)t;
        v = v * (1.0f - s);                                   // detached reset (fwd)
    }
    out_hard[b] = (float)imax;
    out_soft[b] = sum_st / (sum_s + 1e-6f);
}

// ---------------------------------------------------------------------------
extern "C" void kernel_launch(void* const* d_in, const int* in_sizes, int n_in,
                              void* d_out, int out_size, void* d_ws, size_t ws_size,
                              hipStream_t stream)
{
    const float* xa   = (const float*)d_in[0];
    const float* xb   = (const float*)d_in[1];
    const float* W1Aw = (const float*)d_in[2];
    const float* W1Ab = (const float*)d_in[3];
    const float* W2Aw = (const float*)d_in[4];
    const float* W2Ab = (const float*)d_in[5];
    const float* W1Bw = (const float*)d_in[6];
    const float* W1Bb = (const float*)d_in[7];
    const float* W2Bw = (const float*)d_in[8];
    const float* W2Bb = (const float*)d_in[9];

    float* out = (float*)d_out;
    const size_t n_d2 = (size_t)BB * LH * KK2;   // 4,162,048
    const size_t n_bl = (size_t)BB * LH;         //   520,256

    float* da   = out;
    float* ida  = da   + n_d2;
    float* vala = ida  + n_bl;
    float* lah  = vala + n_bl;
    float* las  = lah  + BB;
    float* db   = las  + BB;
    float* idb  = db   + n_d2;
    float* valb = idb  + n_bl;
    float* lbh  = valb + n_bl;
    float* lbs  = lbh  + BB;

    float* IaT = (float*)d_ws;                   // [LH, BB] transposed soma input
    float* IbT = IaT + (size_t)LH * BB;

    dim3 grid(NTILE, BB, 2);                     // 509 tiles x 64 batch x {A,B}
    dendrite_kernel<<<grid, 32, 0, stream>>>(
        xa, xb, W1Aw, W1Ab, W2Aw, W2Ab, W1Bw, W1Bb, W2Bw, W2Bb,
        da, ida, vala, db, idb, valb, IaT, IbT);

    lif_kernel<<<2, 64, 0, stream>>>(IaT, IbT, lah, las, lbh, lbs);
}